// TPHarmonics_11347303596046
// MI455X (gfx1250) — compile-verified
//
#include <hip/hip_runtime.h>
#include <hip/hip_bf16.h>

typedef float v2f  __attribute__((ext_vector_type(2)));
typedef float v8f  __attribute__((ext_vector_type(8)));

#define KDIM   81
#define KPAD   96
#define KK     6561      // 81*81
#define WPB    8         // waves per block (wave32 -> 256 threads)

__global__ __launch_bounds__(256)
void tpharm_wmma_kernel(const float* __restrict__ coords,
                        float* __restrict__ out, int nrow)
{
    __shared__ float psi[WPB][2][KPAD];

    const int lane = threadIdx.x & 31;
    const int wave = threadIdx.x >> 5;
    const int row  = blockIdx.x * WPB + wave;
    if (row >= nrow) return;                 // wave-uniform

    // ---- zero-pad the Psi arrays (entries 81..95 must be 0 for WMMA tiles)
    {
        float* p = &psi[wave][0][0];
        #pragma unroll
        for (int t = 0; t < 2 * KPAD / 32; ++t)
            p[lane + 32 * t] = 0.0f;
    }
    __syncthreads();

    // ---- Phase 1: real spherical harmonics up to L=8, per m-chain per lane.
    // lanes 0..8  -> vector 1 (half=0), lanes 16..24 -> vector 2 (half=1)
    const int half = lane >> 4;
    const int m    = lane & 15;
    if (m <= 8) {
        const float* c = coords + (size_t)row * 6 + half * 3;
        const float x = c[0], y = c[1], z = c[2];
        const float r2xy = x * x + y * y;
        const float rxy  = sqrtf(r2xy);
        const float r    = sqrtf(r2xy + z * z);
        float cosI = fminf(fmaxf(z / r, -1.0f), 1.0f);       // cos(inclination)
        float sinI = sqrtf(fmaxf(1.0f - cosI * cosI, 0.0f)); // >= 0
        // cos/sin of azimuth, algebraically (atan2-free)
        const float cA = (rxy > 0.0f) ? (x / rxy) : 1.0f;
        const float sA = (rxy > 0.0f) ? (y / rxy) : 0.0f;

        // cos(m*az), sin(m*az) via rotation recurrence
        float cm = 1.0f, sm = 0.0f;
        #pragma unroll
        for (int k = 0; k < 8; ++k)
            if (k < m) {
                const float t = cm * cA - sm * sA;
                sm = sm * cA + cm * sA;
                cm = t;
            }

        // P(m,m) = prod_{k=1..m} -(2k-1) * sinI   (Condon-Shortley phase)
        float Pmm = 1.0f;
        #pragma unroll
        for (int k = 1; k <= 8; ++k)
            if (k <= m) Pmm *= -(2.0f * k - 1.0f) * sinI;

        // R(m) = 0!/(2m)! = 1/(2m)!
        double fact = 1.0;
        #pragma unroll
        for (int k = 2; k <= 16; ++k)
            if (k <= 2 * m) fact *= (double)k;
        float Rl = (float)(1.0 / fact);

        const float inv4pi = 0.07957747154594767f;
        const float sqrt2  = 1.4142135623730951f;
        float* ps = &psi[wave][half][0];

        float Pl = Pmm, Plm1 = 0.0f;
        for (int l = m; l <= 8; ++l) {
            const float Nlm  = sqrtf((2.0f * l + 1.0f) * inv4pi * Rl);
            const float base = Nlm * Pl;
            const int   idx  = l * (l + 1);
            if (m == 0) {
                ps[idx] = base;
            } else {
                ps[idx + m] = sqrt2 * base * cm;   // sqrt(2)*Re(Y)
                ps[idx - m] = sqrt2 * base * sm;   // sqrt(2)*Im(Y)
            }
            // advance recurrences: P(l+1,m), R(l+1)
            const float Pn = ((2.0f * l + 1.0f) * cosI * Pl
                              - (float)(l + m) * Plm1) / (float)(l + 1 - m);
            Plm1 = Pl; Pl = Pn;
            Rl *= (float)(l + 1 - m) / (float)(l + m + 1);
        }
    }
    __syncthreads();

    // ---- Phase 2: outer product via rank-1 WMMA, 6x6 tiles of 16x16.
    const float* p1 = &psi[wave][0][0];
    const float* p2 = &psi[wave][1][0];
    float* orow = out + (size_t)row * KK;
    const int jn = lane & 15;
    const int hi = lane >> 4;

    // Hoisted fragment values. Every lane's LDS address is in-bounds
    // (t*16+jn <= 95 < KPAD), so load unconditionally (plain ds_load,
    // no EXEC masking) and zero the upper half-wave with a cndmask.
    float aval[6], bval[6];
    #pragma unroll
    for (int t = 0; t < 6; ++t) {
        const float va = p1[t * 16 + jn];
        const float vb = p2[t * 16 + jn];
        aval[t] = hi ? 0.0f : va;   // A: K=0 column lives in lanes 0..15
        bval[t] = hi ? 0.0f : vb;   // B: K=0 row    lives in lanes 0..15
    }

    #pragma unroll
    for (int ti = 0; ti < 6; ++ti) {
        #pragma unroll
        for (int tj = 0; tj < 6; ++tj) {
            v2f a; a[0] = aval[ti]; a[1] = 0.0f;
            v2f b; b[0] = bval[tj]; b[1] = 0.0f;
            v8f cz = {};
            v8f d = __builtin_amdgcn_wmma_f32_16x16x4_f32(
                false, a, false, b, (short)0, cz, false, false);

            // C/D layout: VGPR v, lanes 0..15 -> M=v, lanes 16..31 -> M=v+8
            const int  jg  = tj * 16 + jn;
            const bool jok = (jg < KDIM);
            #pragma unroll
            for (int v = 0; v < 8; ++v) {
                const int ig = ti * 16 + v + hi * 8;
                if (jok && ig < KDIM)
                    __builtin_nontemporal_store(d[v], orow + ig * KDIM + jg);
            }
        }
    }
}

extern "C" void kernel_launch(void* const* d_in, const int* in_sizes, int n_in,
                              void* d_out, int out_size, void* d_ws, size_t ws_size,
                              hipStream_t stream)
{
    const float* coords = (const float*)d_in[0];
    float* out = (float*)d_out;
    const int nrow = in_sizes[0] / 6;          // 32768
    const int blocks = (nrow + WPB - 1) / WPB; // 4096
    tpharm_wmma_kernel<<<blocks, WPB * 32, 0, stream>>>(coords, out, nrow);
}